// SSM_cha_36386962932304
// MI455X (gfx1250) — compile-verified
//
#include <hip/hip_runtime.h>
#include <hip/hip_bf16.h>
#include <math.h>

// ---- problem constants ----
#define BATCH   16
#define SEQ     4096          // 64*64
#define DM      128           // d_model == d_inner
#define DTRANK  8
#define DSTATE  2
#define NPROJ   12            // DTRANK + 2*DSTATE
#define OUTC    128
#define NCH     64            // scan chunks per sequence
#define CHT     64            // SEQ / NCH steps per chunk

typedef float v2f __attribute__((ext_vector_type(2)));
typedef float v8f __attribute__((ext_vector_type(8)));

__device__ __forceinline__ float silu_f(float x) {
    return x / (1.0f + __expf(-x));
}
__device__ __forceinline__ float softplus_f(float x) {
    return (x > 20.0f) ? x : log1pf(__expf(x));
}

// -----------------------------------------------------------------------------
// Kernel 1: xz = x @ W_in^T  via V_WMMA_F32_16X16X4_F32
// x is stored as x1[b][c][l]  (K-major), W_in is (256,128) row-major.
// Block = 256 threads = 8 waves; each wave owns a 16(M)x32(N) tile.
// grid.x = (B*L)/16 m-tiles.  Cols 0..127 -> xs_raw, 128..255 -> z.
// -----------------------------------------------------------------------------
__global__ void k_in_proj(const float* __restrict__ x1,
                          const float* __restrict__ W_in,
                          float* __restrict__ xs_raw,
                          float* __restrict__ zbuf) {
    const int m0   = blockIdx.x * 16;
    const int b    = m0 / SEQ;
    const int l0   = m0 % SEQ;
    const int lane = threadIdx.x & 31;
    const int wave = threadIdx.x >> 5;
    const int hi   = lane >> 4;
    const int lo   = lane & 15;
    const int n0   = wave * 32;

    const float* Ab = x1 + (size_t)b * DM * SEQ + (l0 + lo);   // + k*SEQ
    const float* B0 = W_in + (size_t)(n0 + lo) * DM;
    const float* B1 = W_in + (size_t)(n0 + 16 + lo) * DM;

    v8f acc0 = {};
    v8f acc1 = {};
    for (int k0 = 0; k0 < DM; k0 += 4) {
        v2f a;                                   // A: M=lo, K = k0+2*hi{,+1}
        a.x = Ab[(size_t)(k0 + 2 * hi) * SEQ];
        a.y = Ab[(size_t)(k0 + 2 * hi + 1) * SEQ];
        const int kk = k0 + hi;                  // B: vgpr0 K=k0+hi, vgpr1 K=k0+hi+2
        v2f bf0, bf1;
        bf0.x = B0[kk];      bf0.y = B0[kk + 2];
        bf1.x = B1[kk];      bf1.y = B1[kk + 2];
        acc0 = __builtin_amdgcn_wmma_f32_16x16x4_f32(false, a, false, bf0,
                                                     (short)0, acc0, false, false);
        acc1 = __builtin_amdgcn_wmma_f32_16x16x4_f32(false, a, false, bf1,
                                                     (short)0, acc1, false, false);
    }
#pragma unroll
    for (int r = 0; r < 8; ++r) {
        const int mrow = m0 + 8 * hi + r;
        const int e0 = n0 + lo;
        const int e1 = e0 + 16;
        if (e0 < DM) xs_raw[(size_t)mrow * DM + e0]        = acc0[r];
        else         zbuf  [(size_t)mrow * DM + (e0 - DM)] = acc0[r];
        if (e1 < DM) xs_raw[(size_t)mrow * DM + e1]        = acc1[r];
        else         zbuf  [(size_t)mrow * DM + (e1 - DM)] = acc1[r];
    }
}

// -----------------------------------------------------------------------------
// Kernel 2: depthwise causal conv (width 4, left pad 3) + bias + SiLU
// -----------------------------------------------------------------------------
__global__ void k_conv_silu(const float* __restrict__ xs_raw,
                            const float* __restrict__ conv_w,
                            const float* __restrict__ conv_b,
                            float* __restrict__ xs) {
    const size_t idx = (size_t)blockIdx.x * blockDim.x + threadIdx.x;
    const size_t total = (size_t)BATCH * SEQ * DM;
    if (idx >= total) return;
    const int d = (int)(idx & (DM - 1));
    const size_t bl = idx >> 7;
    const int l = (int)(bl & (SEQ - 1));

    float acc = conv_b[d];
    const float w0 = conv_w[d * 4 + 0];
    const float w1 = conv_w[d * 4 + 1];
    const float w2 = conv_w[d * 4 + 2];
    const float w3 = conv_w[d * 4 + 3];
    if (l >= 3) acc += w0 * xs_raw[idx - 3 * (size_t)DM];
    if (l >= 2) acc += w1 * xs_raw[idx - 2 * (size_t)DM];
    if (l >= 1) acc += w2 * xs_raw[idx - 1 * (size_t)DM];
    acc += w3 * xs_raw[idx];
    xs[idx] = silu_f(acc);
}

// -----------------------------------------------------------------------------
// Kernel 3: dbl = xs @ W_xproj^T  (12 outputs per token).
// -----------------------------------------------------------------------------
__global__ void k_xproj(const float* __restrict__ xs,
                        const float* __restrict__ W_xproj,
                        float* __restrict__ dbl) {
    __shared__ float sx[DM];
    const int m = blockIdx.x;
    const int t = threadIdx.x;
    sx[t] = xs[(size_t)m * DM + t];
    __syncthreads();
    if (t < NPROJ) {
        const float* wr = W_xproj + (size_t)t * DM;
        float s = 0.0f;
#pragma unroll 8
        for (int d = 0; d < DM; ++d) s += wr[d] * sx[d];
        dbl[(size_t)m * NPROJ + t] = s;
    }
}

// -----------------------------------------------------------------------------
// Chunked scan, phase 1: per-chunk aggregates.
// For state n over a chunk:  h_out = (prod a)*h_in + h_local
// Each thread = one (b, chunk, d).  grid = B*NCH blocks x 128 threads.
// -----------------------------------------------------------------------------
__global__ void k_scan_partial(const float* __restrict__ xs,
                               const float* __restrict__ dbl,
                               const float* __restrict__ W_dt,
                               const float* __restrict__ b_dt,
                               const float* __restrict__ A_log,
                               float* __restrict__ aggA,
                               float* __restrict__ aggH) {
    const int bc = blockIdx.x;          // b*NCH + chunk
    const int b  = bc >> 6;
    const int ch = bc & (NCH - 1);
    const int d  = threadIdx.x;

    float wdt[DTRANK];
#pragma unroll
    for (int r = 0; r < DTRANK; ++r) wdt[r] = W_dt[d * DTRANK + r];
    const float bdt = b_dt[d];
    const float A0 = -__expf(A_log[d * DSTATE + 0]);
    const float A1 = -__expf(A_log[d * DSTATE + 1]);

    float pA0 = 1.0f, pA1 = 1.0f, h0 = 0.0f, h1 = 0.0f;
    const size_t base = (size_t)b * SEQ + (size_t)ch * CHT;
    for (int t = 0; t < CHT; ++t) {
        const size_t m = base + t;
        const float* db = dbl + m * NPROJ;
        float dt = bdt;
#pragma unroll
        for (int r = 0; r < DTRANK; ++r) dt += wdt[r] * db[r];
        dt = softplus_f(dt);
        const float dA0 = __expf(dt * A0);
        const float dA1 = __expf(dt * A1);
        const float xv  = xs[m * DM + d];
        h0 = dA0 * h0 + dt * db[8] * xv;
        h1 = dA1 * h1 + dt * db[9] * xv;
        pA0 *= dA0;
        pA1 *= dA1;
    }
    const size_t o = ((size_t)bc * DM + d) * DSTATE;
    aggA[o]     = pA0;  aggA[o + 1] = pA1;
    aggH[o]     = h0;   aggH[o + 1] = h1;
}

// -----------------------------------------------------------------------------
// Chunked scan, phase 2: exclusive scan over the NCH chunk aggregates.
// Only 64 serial steps per (b,d).  grid = B x 128 threads.
// -----------------------------------------------------------------------------
__global__ void k_scan_combine(const float* __restrict__ aggA,
                               const float* __restrict__ aggH,
                               float* __restrict__ hstart) {
    const int b = blockIdx.x;
    const int d = threadIdx.x;
    float h0 = 0.0f, h1 = 0.0f;
    for (int ch = 0; ch < NCH; ++ch) {
        const size_t o = (((size_t)b * NCH + ch) * DM + d) * DSTATE;
        hstart[o]     = h0;
        hstart[o + 1] = h1;
        h0 = aggA[o]     * h0 + aggH[o];
        h1 = aggA[o + 1] * h1 + aggH[o + 1];
    }
}

// -----------------------------------------------------------------------------
// Chunked scan, phase 3: re-run each chunk from its start state and emit
// gated y = (h.C + xs*D) * silu(z).  grid = B*NCH blocks x 128 threads.
// -----------------------------------------------------------------------------
__global__ void k_scan_emit(const float* __restrict__ xs,
                            const float* __restrict__ zbuf,
                            const float* __restrict__ dbl,
                            const float* __restrict__ W_dt,
                            const float* __restrict__ b_dt,
                            const float* __restrict__ A_log,
                            const float* __restrict__ D_param,
                            const float* __restrict__ hstart,
                            float* __restrict__ ybuf) {
    const int bc = blockIdx.x;
    const int b  = bc >> 6;
    const int ch = bc & (NCH - 1);
    const int d  = threadIdx.x;

    float wdt[DTRANK];
#pragma unroll
    for (int r = 0; r < DTRANK; ++r) wdt[r] = W_dt[d * DTRANK + r];
    const float bdt = b_dt[d];
    const float A0 = -__expf(A_log[d * DSTATE + 0]);
    const float A1 = -__expf(A_log[d * DSTATE + 1]);
    const float Dp = D_param[d];

    const size_t oh = ((size_t)bc * DM + d) * DSTATE;
    float h0 = hstart[oh];
    float h1 = hstart[oh + 1];

    const size_t base = (size_t)b * SEQ + (size_t)ch * CHT;
    for (int t = 0; t < CHT; ++t) {
        const size_t m = base + t;
        const float* db = dbl + m * NPROJ;
        float dt = bdt;
#pragma unroll
        for (int r = 0; r < DTRANK; ++r) dt += wdt[r] * db[r];
        dt = softplus_f(dt);
        const float xv = xs[m * DM + d];
        h0 = __expf(dt * A0) * h0 + dt * db[8] * xv;
        h1 = __expf(dt * A1) * h1 + dt * db[9] * xv;
        float yv = h0 * db[10] + h1 * db[11] + xv * Dp;
        const float zv = zbuf[m * DM + d];
        yv *= silu_f(zv);
        ybuf[m * DM + d] = yv;
    }
}

// -----------------------------------------------------------------------------
// Kernel 5: out = LayerNorm(y @ W_out^T) -> (B, OUT_C, H, W) layout.
// Block = 128 threads = 4 waves; 16 rows x 128 cols via WMMA; fused LN.
// -----------------------------------------------------------------------------
__global__ void k_out_ln(const float* __restrict__ ybuf,
                         const float* __restrict__ W_out,
                         const float* __restrict__ ln_g,
                         const float* __restrict__ ln_b,
                         float* __restrict__ out) {
    __shared__ float tile[16][OUTC + 1];
    __shared__ float smu[16], srs[16];

    const int m0   = blockIdx.x * 16;
    const int lane = threadIdx.x & 31;
    const int wave = threadIdx.x >> 5;
    const int hi   = lane >> 4;
    const int lo   = lane & 15;
    const int n0   = wave * 32;

    const float* B0 = W_out + (size_t)(n0 + lo) * DM;
    const float* B1 = W_out + (size_t)(n0 + 16 + lo) * DM;

    v8f acc0 = {};
    v8f acc1 = {};
    for (int k0 = 0; k0 < DM; k0 += 4) {
        v2f a;
        a.x = ybuf[(size_t)(m0 + lo) * DM + k0 + 2 * hi];
        a.y = ybuf[(size_t)(m0 + lo) * DM + k0 + 2 * hi + 1];
        const int kk = k0 + hi;
        v2f bf0, bf1;
        bf0.x = B0[kk];      bf0.y = B0[kk + 2];
        bf1.x = B1[kk];      bf1.y = B1[kk + 2];
        acc0 = __builtin_amdgcn_wmma_f32_16x16x4_f32(false, a, false, bf0,
                                                     (short)0, acc0, false, false);
        acc1 = __builtin_amdgcn_wmma_f32_16x16x4_f32(false, a, false, bf1,
                                                     (short)0, acc1, false, false);
    }
#pragma unroll
    for (int r = 0; r < 8; ++r) {
        const int row = 8 * hi + r;
        tile[row][n0 + lo]      = acc0[r];
        tile[row][n0 + 16 + lo] = acc1[r];
    }
    __syncthreads();

    const int t = threadIdx.x;          // 0..127 == output channel o
    if (t < 16) {
        float s = 0.0f;
        for (int o = 0; o < OUTC; ++o) s += tile[t][o];
        const float mu = s * (1.0f / OUTC);
        float v = 0.0f;
        for (int o = 0; o < OUTC; ++o) {
            const float dd = tile[t][o] - mu;
            v += dd * dd;
        }
        smu[t] = mu;
        srs[t] = rsqrtf(v * (1.0f / OUTC) + 1e-5f);
    }
    __syncthreads();

    const int b  = m0 / SEQ;
    const int l0 = m0 % SEQ;
    const float g  = ln_g[t];
    const float be = ln_b[t];
    float* ob = out + (size_t)b * OUTC * SEQ + (size_t)t * SEQ + l0;
#pragma unroll
    for (int r = 0; r < 16; ++r) {
        ob[r] = (tile[r][t] - smu[r]) * srs[r] * g + be;
    }
}

// -----------------------------------------------------------------------------
extern "C" void kernel_launch(void* const* d_in, const int* in_sizes, int n_in,
                              void* d_out, int out_size, void* d_ws, size_t ws_size,
                              hipStream_t stream) {
    (void)in_sizes; (void)n_in; (void)out_size; (void)ws_size;
    const float* x1      = (const float*)d_in[0];
    const float* W_in    = (const float*)d_in[1];
    const float* conv_w  = (const float*)d_in[2];
    const float* conv_b  = (const float*)d_in[3];
    const float* W_xproj = (const float*)d_in[4];
    const float* W_dt    = (const float*)d_in[5];
    const float* b_dt    = (const float*)d_in[6];
    const float* A_log   = (const float*)d_in[7];
    const float* D_param = (const float*)d_in[8];
    const float* W_out   = (const float*)d_in[9];
    const float* ln_g    = (const float*)d_in[10];
    const float* ln_b    = (const float*)d_in[11];
    float* out = (float*)d_out;

    float* ws = (float*)d_ws;
    const size_t SZ  = (size_t)BATCH * SEQ * DM;            // 8,388,608 floats
    const size_t AGG = (size_t)BATCH * NCH * DM * DSTATE;   // 262,144 floats
    float* xs_raw = ws;                     // pre-conv x path (reused as y)
    float* zbuf   = ws + SZ;                // gate path
    float* xs     = ws + 2 * SZ;            // post conv+SiLU
    float* dbl    = ws + 3 * SZ;            // (B*L, 12)
    float* aggA   = dbl  + (size_t)BATCH * SEQ * NPROJ;
    float* aggH   = aggA + AGG;
    float* hstart = aggH + AGG;
    float* ybuf   = xs_raw;                 // safe reuse after scan inputs read

    const int MTILES = (BATCH * SEQ) / 16;  // 4096

    k_in_proj <<<MTILES, 256, 0, stream>>>(x1, W_in, xs_raw, zbuf);
    {
        const size_t total = (size_t)BATCH * SEQ * DM;
        k_conv_silu<<<(unsigned)((total + 255) / 256), 256, 0, stream>>>(
            xs_raw, conv_w, conv_b, xs);
    }
    k_xproj       <<<BATCH * SEQ, DM, 0, stream>>>(xs, W_xproj, dbl);
    k_scan_partial<<<BATCH * NCH, DM, 0, stream>>>(xs, dbl, W_dt, b_dt, A_log,
                                                   aggA, aggH);
    k_scan_combine<<<BATCH, DM, 0, stream>>>(aggA, aggH, hstart);
    k_scan_emit   <<<BATCH * NCH, DM, 0, stream>>>(xs, zbuf, dbl, W_dt, b_dt,
                                                   A_log, D_param, hstart, ybuf);
    k_out_ln      <<<MTILES, DM, 0, stream>>>(ybuf, W_out, ln_g, ln_b, out);
}